// LateralInhibitionLIFCell_26972394619167
// MI455X (gfx1250) — compile-verified
//
#include <hip/hip_runtime.h>

// -----------------------------------------------------------------------------
// LIF step + per-row winner-take-all lateral inhibition, MI455X (gfx1250).
//
// Memory-bound: 3 loads + 3 stores * 4B * 4096*8192 = ~805 MB  =>  ~35 us at
// 23.3 TB/s. No matmul exists in this op, so no WMMA path applies; the design
// targets peak HBM streaming:
//   * one 256-thread block (8 wave32 waves) per row
//   * b128 vector memory ops with non-temporal (TH=NT) policy (touch-once data,
//     keep the 192MB L2 clean)
//   * argmax via monotone u64 key + wave32 butterfly shuffle + one LDS
//     atomicMax(u64) per wave (DS_MAX_RTN_U64), first-index tie-break encoded
//     as (0xFFFFFFFF - col) in the low dword
//   * pass 2 needs no reload: winner's v_out is exactly V_RESET (it spiked),
//     non-winners are the constant -5.0, no-spike rows replay v_after kept in
//     VGPRs.
// -----------------------------------------------------------------------------

typedef __attribute__((ext_vector_type(4))) float v4f;

#define N_COLS   8192
#define THREADS  256
#define VPT      (N_COLS / (THREADS * 4))   // 8 float4 chunks per thread

#define V_TH_C    1.0f
#define V_RESET_C 0.0f
#define INHIB_C  -5.0f

__global__ __launch_bounds__(THREADS)
void lif_wta_kernel(const float* __restrict__ x,
                    const float* __restrict__ v,
                    const float* __restrict__ cur,
                    float* __restrict__ z_out,
                    float* __restrict__ v_out,
                    float* __restrict__ i_out)
{
    __shared__ unsigned long long s_key;
    const int t = threadIdx.x;
    if (t == 0) s_key = 0ull;

    const long long base = (long long)blockIdx.x * N_COLS;
    const v4f* x4  = (const v4f*)(x   + base);
    const v4f* v4p = (const v4f*)(v   + base);
    const v4f* i4  = (const v4f*)(cur + base);
    v4f* z4 = (v4f*)(z_out + base);
    v4f* o4 = (v4f*)(v_out + base);
    v4f* n4 = (v4f*)(i_out + base);

    float va[VPT * 4];                 // v_after, kept in VGPRs for pass 2
    unsigned long long best = 0ull;    // (score_bits << 32) | (0xFFFFFFFF - col)

    // ---- pass 1: elementwise LIF, emit z and i_new, track local argmax ----
#pragma unroll
    for (int k = 0; k < VPT; ++k) {
        const int vecIdx = k * THREADS + t;       // coalesced, strided
        v4f xx = __builtin_nontemporal_load(x4  + vecIdx);
        v4f vv = __builtin_nontemporal_load(v4p + vecIdx);
        v4f ii = __builtin_nontemporal_load(i4  + vecIdx);
        v4f zz, in4;
#pragma unroll
        for (int c = 0; c < 4; ++c) {
            const float inew = ii[c] + 0.5f * (xx[c] - ii[c]);
            const float vnew = vv[c] + 0.5f * (inew - vv[c]);
            const bool spike = vnew >= V_TH_C;
            in4[c]        = inew;
            zz[c]         = spike ? 1.0f : 0.0f;
            va[k * 4 + c] = spike ? V_RESET_C : vnew;
            if (spike) {
                // vnew >= 1.0 > 0  =>  f32 bits are order-preserving as u32
                const unsigned sb = __float_as_uint(vnew);
                const unsigned j  = (unsigned)(vecIdx * 4 + c);
                const unsigned long long key =
                    ((unsigned long long)sb << 32) |
                    (unsigned long long)(0xFFFFFFFFu - j);
                best = (key > best) ? key : best;
            }
        }
        __builtin_nontemporal_store(zz,  z4 + vecIdx);
        __builtin_nontemporal_store(in4, n4 + vecIdx);
    }

    // ---- wave32 butterfly max-reduction of the 64-bit key ----
#pragma unroll
    for (int off = 16; off > 0; off >>= 1) {
        const unsigned lo  = (unsigned)best;
        const unsigned hi  = (unsigned)(best >> 32);
        const unsigned olo = __shfl_xor(lo, off, 32);
        const unsigned ohi = __shfl_xor(hi, off, 32);
        const unsigned long long other =
            ((unsigned long long)ohi << 32) | (unsigned long long)olo;
        best = (other > best) ? other : best;
    }

    __syncthreads();                                  // s_key init visible
    if ((t & 31) == 0 && best != 0ull)
        atomicMax(&s_key, best);                      // DS u64 max, 8/block
    __syncthreads();

    // ---- pass 2: write v_out (branch is uniform across the block) ----
    const unsigned long long key = s_key;
    if (key == 0ull) {
        // no spikes in this row: v_out = v_after (== v_new)
#pragma unroll
        for (int k = 0; k < VPT; ++k) {
            const int vecIdx = k * THREADS + t;
            v4f out;
#pragma unroll
            for (int c = 0; c < 4; ++c) out[c] = va[k * 4 + c];
            __builtin_nontemporal_store(out, o4 + vecIdx);
        }
    } else {
        // winner spiked => its v_after is V_RESET; all others are inhibited
        const unsigned winner = 0xFFFFFFFFu - (unsigned)(key & 0xFFFFFFFFull);
#pragma unroll
        for (int k = 0; k < VPT; ++k) {
            const int vecIdx = k * THREADS + t;
            v4f out;
#pragma unroll
            for (int c = 0; c < 4; ++c)
                out[c] = ((unsigned)(vecIdx * 4 + c) == winner) ? V_RESET_C
                                                                : INHIB_C;
            __builtin_nontemporal_store(out, o4 + vecIdx);
        }
    }
}

extern "C" void kernel_launch(void* const* d_in, const int* in_sizes, int n_in,
                              void* d_out, int out_size, void* d_ws, size_t ws_size,
                              hipStream_t stream) {
    const float* x = (const float*)d_in[0];
    const float* v = (const float*)d_in[1];
    const float* i = (const float*)d_in[2];

    const long long BN = (long long)in_sizes[0];     // B * N
    const int B = (int)(BN / N_COLS);                // rows (N fixed at 8192)

    float* z_out = (float*)d_out;                    // return order: z, v, i
    float* v_out = z_out + BN;
    float* i_out = z_out + 2 * BN;

    lif_wta_kernel<<<dim3(B), dim3(THREADS), 0, stream>>>(x, v, i,
                                                          z_out, v_out, i_out);
}